// IDWT2D_25580825215626
// MI455X (gfx1250) — compile-verified
//
#include <hip/hip_runtime.h>

// CDNA5 / gfx1250 wave32 WMMA implementation of batched 2D inverse db4 DWT.
// V_WMMA_F32_16X16X4_F32 (exact fp32) over the 16x16 block-Toeplitz structure
// of the upsample+convolve reconstruction operator.
//
// Key structure: output row tile r=0..15 contracts only input rows
// i in [r/2, r/2+3], so i <= 10: columns 11..15 of the G tile are exactly
// zero. The contraction therefore needs only K=12 (3 chunks of 4), i.e.
// 18 WMMAs and 24 loads per tile instead of 24/32 — 25% less read traffic
// on a bandwidth-bound kernel.
//
// Out-of-bounds input rows/cols are index-CLAMPED (not zero-filled): any
// stored output element only contracts clamped (finite garbage) values
// against exactly-zero G coefficients, preserving results bit-exactly while
// keeping all loads branchless. One wave per workgroup keeps tile indices
// scalar (SALU addressing, scalar branch for the store boundary path).

typedef __attribute__((ext_vector_type(2))) float v2f;
typedef __attribute__((ext_vector_type(8))) float v8f;

#define IN_H   256
#define IN_W   256
#define OUT_H  506
#define OUT_W  506

// pywt db4 reconstruction filters
__constant__ float c_rec_lo[8] = {
    0.23037781330885523f,  0.7148465705525415f,  0.6308807679295904f,
   -0.02798376941698385f, -0.18703481171888114f, 0.030841381835986965f,
    0.032883011666982945f, -0.010597401784997278f};
__constant__ float c_rec_hi[8] = {
   -0.010597401784997278f, -0.032883011666982945f, 0.030841381835986965f,
    0.18703481171888114f,  -0.02798376941698385f,  -0.6308807679295904f,
    0.7148465705525415f,   -0.23037781330885523f};

__device__ __forceinline__ int imin(int a, int b) { return a < b ? a : b; }

// Block-Toeplitz reconstruction tile coefficient: out-row r, in-row i.
// Nonzero iff 0 <= i - r/2 <= 3  (so i <= 10 for r <= 15).
__device__ __forceinline__ float gcoef(const float* rec, int r, int i) {
  int j = i - (r >> 1);
  return (j >= 0 && j < 4) ? rec[6 - 2 * j + (r & 1)] : 0.0f;
}

// D = A(16x4) * B(4x16) + C(16x16), all fp32.
// A: lanes 0-15 row M=lane {v0=K0,v1=K1}; lanes 16-31 {v0=K2,v1=K3}
// B: lanes 0-15 col N=lane {v0=K0,v1=K1}; lanes 16-31 {v0=K2,v1=K3}
// C/D: vgpr j -> row j (lanes 0-15) / row j+8 (lanes 16-31), col = lane%16
__device__ __forceinline__ v8f wmma4(v2f a, v2f b, v8f c) {
  return __builtin_amdgcn_wmma_f32_16x16x4_f32(
      /*neg_a=*/false, a, /*neg_b=*/false, b,
      /*c_mod=*/(short)0, c, /*reuse_a=*/false, /*reuse_b=*/false);
}

__global__ __launch_bounds__(32) void idwt2_db4_wmma(
    const float* __restrict__ LL, const float* __restrict__ LH,
    const float* __restrict__ HL, const float* __restrict__ HH,
    float* __restrict__ OUT) {
  __shared__ float ldsLo[16 * 17];  // stride 17: bank-conflict-free transpose
  __shared__ float ldsHi[16 * 17];

  const int lane = threadIdx.x;   // 0..31 (single wave per block)
  const int h    = lane >> 4;     // half-wave select (K pairing / row+8)
  const int c    = lane & 15;     // M (A rows) == N (B cols) index

  const int s     = blockIdx.x >> 5;   // output row-tile 0..31  (scalar)
  const int t     = blockIdx.x & 31;   // output col-tile 0..31  (scalar)
  const int slice = blockIdx.y;        // b*64 + ch, 0..255      (scalar)

  const size_t in_off = (size_t)slice * (IN_H * IN_W);
  const float* __restrict__ ll = LL + in_off;
  const float* __restrict__ lh = LH + in_off;
  const float* __restrict__ hl = HL + in_off;
  const float* __restrict__ hh = HH + in_off;

  // G chunks (K=12: chunks 0..2; chunk 3 is identically zero).
  // Serve as A (vertical pass) AND as B = G^T chunks (horizontal pass).
  v2f glo[3], ghi[3];
#pragma unroll
  for (int kk = 0; kk < 3; ++kk) {
    const int i0 = 4 * kk + 2 * h;
    glo[kk].x = gcoef(c_rec_lo, c, i0);
    glo[kk].y = gcoef(c_rec_lo, c, i0 + 1);
    ghi[kk].x = gcoef(c_rec_hi, c, i0);
    ghi[kk].y = gcoef(c_rec_hi, c, i0 + 1);
  }

  // Clamped input column (garbage beyond 255 hits zero coefficients only).
  const int col = imin(8 * t + c, IN_W - 1);

  // ---------- vertical pass: Vlo = Glo*LL + Ghi*LH ; Vhi = Glo*HL + Ghi*HH
  v8f vlo = {0.f, 0.f, 0.f, 0.f, 0.f, 0.f, 0.f, 0.f};
  v8f vhi = vlo;

#pragma unroll
  for (int kk = 0; kk < 3; ++kk) {
    const int r0 = 8 * s + 4 * kk + 2 * h;
    const int o0 = imin(r0, IN_H - 1) * IN_W + col;      // clamped rows
    const int o1 = imin(r0 + 1, IN_H - 1) * IN_W + col;
    v2f b;
    b.x = ll[o0];  b.y = ll[o1];
    vlo = wmma4(glo[kk], b, vlo);
    b.x = lh[o0];  b.y = lh[o1];
    vlo = wmma4(ghi[kk], b, vlo);
    b.x = hl[o0];  b.y = hl[o1];
    vhi = wmma4(glo[kk], b, vhi);
    b.x = hh[o0];  b.y = hh[o1];
    vhi = wmma4(ghi[kk], b, vhi);
  }

  // ---------- re-layout V tiles (C/D layout -> A layout) through LDS
#pragma unroll
  for (int j = 0; j < 8; ++j) {
    ldsLo[(j + 8 * h) * 17 + c] = vlo[j];
    ldsHi[(j + 8 * h) * 17 + c] = vhi[j];
  }
  __syncthreads();  // single-wave workgroup: NOP + ordering fence

  // ---------- horizontal pass: Out = Vlo*Glo^T + Vhi*Ghi^T (K=12)
  v8f out = {0.f, 0.f, 0.f, 0.f, 0.f, 0.f, 0.f, 0.f};
#pragma unroll
  for (int kk = 0; kk < 3; ++kk) {
    const int k0 = 4 * kk + 2 * h;
    v2f a;
    a.x = ldsLo[c * 17 + k0];
    a.y = ldsLo[c * 17 + k0 + 1];
    out = wmma4(a, glo[kk], out);
    a.x = ldsHi[c * 17 + k0];
    a.y = ldsHi[c * 17 + k0 + 1];
    out = wmma4(a, ghi[kk], out);
  }

  // ---------- store (506 = 31*16 + 10): scalar-branch fast path
  float* __restrict__ op = OUT + (size_t)slice * (OUT_H * OUT_W);
  const int orow0 = 16 * s + 8 * h;
  const int ocol  = 16 * t + c;

  if (s < 31 && t < 31) {           // interior tile: fully in-bounds (94%)
#pragma unroll
    for (int j = 0; j < 8; ++j)
      op[(size_t)(orow0 + j) * OUT_W + ocol] = out[j];
  } else {                           // boundary tile: guarded stores
    if (ocol < OUT_W) {
#pragma unroll
      for (int j = 0; j < 8; ++j) {
        const int orow = orow0 + j;
        if (orow < OUT_H)
          op[(size_t)orow * OUT_W + ocol] = out[j];
      }
    }
  }
}

extern "C" void kernel_launch(void* const* d_in, const int* in_sizes, int n_in,
                              void* d_out, int out_size, void* d_ws, size_t ws_size,
                              hipStream_t stream) {
  (void)in_sizes; (void)n_in; (void)out_size; (void)d_ws; (void)ws_size;
  const float* LL = (const float*)d_in[0];
  const float* LH = (const float*)d_in[1];
  const float* HL = (const float*)d_in[2];
  const float* HH = (const float*)d_in[3];
  float* OUT = (float*)d_out;

  // 256 slices (4*64); 32x32 tiles per slice; one wave (16x16 out tile) each
  dim3 grid(1024, 256, 1);
  dim3 block(32, 1, 1);
  idwt2_db4_wmma<<<grid, block, 0, stream>>>(LL, LH, HL, HH, OUT);
}